// kernel_generated_4_21887153341275
// MI455X (gfx1250) — compile-verified
//
#include <hip/hip_runtime.h>

typedef float v2f __attribute__((ext_vector_type(2)));
typedef float v8f __attribute__((ext_vector_type(8)));

#define H28 28
#define P784 784            // 28*28
#define C32 32
#define ROW588 588          // 28*21
#define KPAD 64             // K = 3*21 = 63, padded to 64
#define MOUT 256

// ---- Stage 0: t3 with width-roll (+1): t3r[c,h,w] = x[c,h,(w-1)%28] + x[c+32,h,(w-1)%28]
__global__ void k_t3(const float* __restrict__ x, float* __restrict__ t3r) {
    int tid = blockIdx.x * blockDim.x + threadIdx.x;
    if (tid >= C32 * P784) return;
    int c = tid / P784, rem = tid % P784;
    int h = rem / H28, w = rem % H28;
    int wsrc = (w + H28 - 1) % H28;          // roll by +1
    int src = c * P784 + h * H28 + wsrc;
    t3r[tid] = x[src] + x[src + C32 * P784];
}

// ---- Stage 1: t5 (width conv, 21 feats), stored padded over h: t5p has 30 rows of 588
// t5p[(h+1)*588 + w*21 + kc] = sum_{c,j} t3r[c,h,w+j-3] * w1[c*147 + j*21 + kc]
__global__ void k_t5(const float* __restrict__ t3r, const float* __restrict__ w1,
                     float* __restrict__ t5p) {
    int tid = blockIdx.x * blockDim.x + threadIdx.x;
    if (tid >= 30 * ROW588) return;
    int row = tid / ROW588;
    int rem = tid % ROW588;
    if (row == 0 || row == 29) { t5p[tid] = 0.0f; return; }   // zero h-padding rows
    int h = row - 1;
    int w = rem / 21, kc = rem % 21;
    float acc = 0.0f;
    for (int c = 0; c < C32; ++c) {
        const float* t3row = t3r + c * P784 + h * H28;
        const float* w1row = w1 + c * 147 + kc;
#pragma unroll
        for (int j = 0; j < 7; ++j) {
            int ww = w + j - 3;
            float v = (ww >= 0 && ww < H28) ? t3row[ww] : 0.0f;
            acc += v * w1row[j * 21];
        }
    }
    t5p[tid] = acc;
}

// ---- Pack A: w0 (256 x 63) -> w0p (256 x 64), last K column zero
__global__ void k_packA(const float* __restrict__ w0, float* __restrict__ w0p) {
    int tid = blockIdx.x * blockDim.x + threadIdx.x;
    if (tid >= MOUT * KPAD) return;
    int i = tid >> 6, k = tid & 63;
    w0p[tid] = (k < 63) ? w0[i * 63 + k] : 0.0f;
}

// ---- Pack B: im2col over h of t5p -> Bmat stored p-major: Bm[p][k], (784 x 64)
// p-major means each GEMM lane's (k0, k0+1) pair is contiguous -> one b64 load.
__global__ void k_packB(const float* __restrict__ t5p, float* __restrict__ Bm) {
    int tid = blockIdx.x * blockDim.x + threadIdx.x;
    if (tid >= P784 * KPAD) return;
    int p = tid >> 6, k = tid & 63;
    float v = 0.0f;
    if (k < 63) {
        int j = k / 21, kc = k % 21;
        int h = p / H28, w = p % H28;
        v = t5p[(h + j) * ROW588 + w * 21 + kc];   // rows pre-padded, no bounds check
    }
    Bm[tid] = v;
}

// ---- Stage 2 GEMM via WMMA f32: D(256x784) = A(256x64) * B(64x784)
// One wave (32 lanes) per 16x16 tile. grid = (16, 49), block = 32.
// A fragment (16x4 f32, ISA layout): lane = (k/2)*16 + m, vgpr = k&1.
// B fragment (4x16 f32, symmetric): lane = (k/2)*16 + n, vgpr = k&1.
// C/D (16x16 f32): vgpr r, lanes 0-15 -> M=r, lanes 16-31 -> M=r+8, N=lane%16.
__global__ void __launch_bounds__(32)
k_gemm(const float* __restrict__ A, const float* __restrict__ B, float* __restrict__ D) {
    const int lane = threadIdx.x;
    const int t    = lane & 15;       // m (for A) / n (for B & D)
    const int kh   = lane >> 4;       // which k-pair half
    const int mi   = blockIdx.x;      // 0..15   (M tile)
    const int ni   = blockIdx.y;      // 0..48   (N tile)

    const float* Arow = A + (size_t)(mi * 16 + t) * KPAD;  // A[m=t][.]
    const float* Brow = B + (size_t)(ni * 16 + t) * KPAD;  // B[.][n=t], p-major

    v8f c = {0.f, 0.f, 0.f, 0.f, 0.f, 0.f, 0.f, 0.f};
#pragma unroll
    for (int kk = 0; kk < 16; ++kk) {
        const int k0 = kk * 4 + kh * 2;
        v2f a = *(const v2f*)(Arow + k0);   // contiguous pair -> global_load_b64
        v2f b = *(const v2f*)(Brow + k0);   // contiguous pair -> global_load_b64
        c = __builtin_amdgcn_wmma_f32_16x16x4_f32(
                /*neg_a=*/false, a, /*neg_b=*/false, b,
                /*c_mod=*/(short)0, c, /*reuse_a=*/false, /*reuse_b=*/false);
    }

#pragma unroll
    for (int r = 0; r < 8; ++r) {
        const int m = r + kh * 8;
        D[(size_t)(mi * 16 + m) * P784 + ni * 16 + t] = c[r];
    }
}

extern "C" void kernel_launch(void* const* d_in, const int* in_sizes, int n_in,
                              void* d_out, int out_size, void* d_ws, size_t ws_size,
                              hipStream_t stream) {
    const float* x  = (const float*)d_in[0];   // (1,64,28,28)
    const float* w0 = (const float*)d_in[1];   // (256,3,21)
    const float* w1 = (const float*)d_in[2];   // (32,7,21)
    float* y = (float*)d_out;                  // (1,256,28,28)

    // Workspace layout (floats)
    float* ws  = (float*)d_ws;
    float* t3r = ws;                  // 32*784          = 25088
    float* t5p = ws + 25088;          // 30*588          = 17640
    float* w0p = ws + 42728;          // 256*64          = 16384
    float* Bm  = ws + 59112;          // 784*64          = 50176
                                      // total 109288 floats (~437 KB)

    k_t3   <<<(C32 * P784 + 255) / 256, 256, 0, stream>>>(x, t3r);
    k_t5   <<<(30 * ROW588 + 255) / 256, 256, 0, stream>>>(t3r, w1, t5p);
    k_packA<<<(MOUT * KPAD) / 256,      256, 0, stream>>>(w0, w0p);
    k_packB<<<(P784 * KPAD) / 256,      256, 0, stream>>>(t5p, Bm);
    k_gemm <<<dim3(16, 49), 32, 0, stream>>>(w0p, Bm, y);
}